// SparseDeepHaloChoice_33045478375777
// MI455X (gfx1250) — compile-verified
//
#include <hip/hip_runtime.h>
#include <math.h>

typedef __attribute__((ext_vector_type(8))) float v8f;
typedef __attribute__((ext_vector_type(2))) float v2f;

#define NB 2048
#define NI 1024
#define NF 64
#define NH 32
#define THREADS 256
#define WAVES 8
#define TPW 8  // item-tiles per wave (64 tiles / 8 waves)

struct LdsBuf {
  float z0[NI * NH];          // 128 KB: per-batch embeddings, LDS-resident
  float ws[WAVES][16 * NH];   // 16 KB: per-wave C-layout <-> A-layout transpose scratch
  float maskf[NI];
  float halo[NI];
  float s0[NH];
  float sm0[NH];
  float zbp0[NH];
  float zbp1[NH];
  float cnt;
};

__device__ __forceinline__ v8f z8() {
  v8f v = {0.f, 0.f, 0.f, 0.f, 0.f, 0.f, 0.f, 0.f};
  return v;
}

// D = A(16x4 f32) * B(4x16 f32) + C(16x16 f32)
__device__ __forceinline__ v8f wmma4(v2f a, v2f b, v8f c) {
  return __builtin_amdgcn_wmma_f32_16x16x4_f32(false, a, false, b, (short)0, c,
                                               false, false);
}

__global__ __launch_bounds__(THREADS, 1) void halo_fused(
    const float* __restrict__ X, const int* __restrict__ mask,
    const int* __restrict__ mids, const float* __restrict__ We1,
    const float* __restrict__ be1, const float* __restrict__ We2,
    const float* __restrict__ be2, const float* __restrict__ Wp,
    const float* __restrict__ bp, const float* __restrict__ Wm,
    const float* __restrict__ bm, const float* __restrict__ Wh,
    const float* __restrict__ bh, const float* __restrict__ xi,
    float* __restrict__ out) {
  __shared__ LdsBuf lds;
  const int tid = threadIdx.x;
  const int l = tid & 31;        // lane in wave32
  const int w = tid >> 5;        // wave id (0..7)
  const int b = blockIdx.x;      // batch
  const int col = l & 15;        // N-column (and A-matrix row) for this lane
  const int khalf = l >> 4;      // lanes 16-31 hold K+2/K+3 of A, M+8 of C/D
  const int rowoff = khalf << 3; // C/D row offset

  if (tid < NH) { lds.s0[tid] = 0.f; lds.sm0[tid] = 0.f; }
  if (tid == NH) lds.cnt = 0.f;
  __syncthreads();

  // ---- mask -> LDS (as float) + masked count ----
  float cpart = 0.f;
  for (int n = tid; n < NI; n += THREADS) {
    float mv = (float)mask[(size_t)b * NI + n];
    lds.maskf[n] = mv;
    cpart += mv;
  }
#pragma unroll
  for (int o = 16; o > 0; o >>= 1) cpart += __shfl_xor(cpart, o, 32);
  if (l == 0) atomicAdd(&lds.cnt, cpart);

  // ==== Phase A: z0 = relu(X@We1+be1)@We2+be2, streamed from HBM once ====
  {
    v2f bE1[16][2];  // B-fragments of We1 (64x32), kept in VGPRs for all tiles
#pragma unroll
    for (int ks = 0; ks < 16; ++ks) {
      const int kr = ks * 4 + khalf * 2;
#pragma unroll
      for (int nt = 0; nt < 2; ++nt) {
        const int c = nt * 16 + col;
        v2f t;
        t[0] = We1[kr * NH + c];
        t[1] = We1[(kr + 1) * NH + c];
        bE1[ks][nt] = t;
      }
    }
    v2f bE2[8][2];  // B-fragments of We2 (32x32)
#pragma unroll
    for (int ks = 0; ks < 8; ++ks) {
      const int kr = ks * 4 + khalf * 2;
#pragma unroll
      for (int nt = 0; nt < 2; ++nt) {
        const int c = nt * 16 + col;
        v2f t;
        t[0] = We2[kr * NH + c];
        t[1] = We2[(kr + 1) * NH + c];
        bE2[ks][nt] = t;
      }
    }
    const float vbe1[2] = {be1[col], be1[16 + col]};
    const float vbe2[2] = {be2[col], be2[16 + col]};
    float* ws = lds.ws[w];

    for (int i = 0; i < TPW; ++i) {
      const int ib = (w * TPW + i) * 16;
      const float* Xrow = X + ((size_t)b * NI + ib + col) * NF + khalf * 2;
      if (i + 1 < TPW) __builtin_prefetch(Xrow + 16 * NF, 0, 0);

      v8f a0 = z8(), a1 = z8();
#pragma unroll
      for (int ks = 0; ks < 16; ++ks) {  // K = 64
        v2f af = *(const v2f*)(Xrow + ks * 4);
        a0 = wmma4(af, bE1[ks][0], a0);
        a1 = wmma4(af, bE1[ks][1], a1);
      }
      // bias + relu, transpose via per-wave LDS scratch (C-layout -> A-layout)
#pragma unroll
      for (int r = 0; r < 8; ++r) {
        float x0 = a0[r] + vbe1[0];
        float x1 = a1[r] + vbe1[1];
        ws[(r + rowoff) * NH + col] = x0 > 0.f ? x0 : 0.f;
        ws[(r + rowoff) * NH + 16 + col] = x1 > 0.f ? x1 : 0.f;
      }
      v8f c0 = z8(), c1 = z8();
#pragma unroll
      for (int ks = 0; ks < 8; ++ks) {  // K = 32
        v2f af = *(const v2f*)&ws[col * NH + ks * 4 + khalf * 2];
        c0 = wmma4(af, bE2[ks][0], c0);
        c1 = wmma4(af, bE2[ks][1], c1);
      }
      float* zrow = &lds.z0[(size_t)ib * NH];
#pragma unroll
      for (int r = 0; r < 8; ++r) {
        zrow[(r + rowoff) * NH + col] = c0[r] + vbe2[0];
        zrow[(r + rowoff) * NH + 16 + col] = c1[r] + vbe2[1];
      }
    }
  }
  __syncthreads();

  // ==== Phase B: s0[h] = sum_n m[n]*z0[n,h] (LDS-only) ====
  {
    float p = 0.f;
    for (int n = w; n < NI; n += WAVES) p += lds.maskf[n] * lds.z0[n * NH + l];
    atomicAdd(&lds.s0[l], p);
  }

  // B-fragments of W_mod[0] and W_mod[1] (32x32 each), live through C+E
  v2f bM0[8][2], bM1[8][2];
#pragma unroll
  for (int ks = 0; ks < 8; ++ks) {
    const int kr = ks * 4 + khalf * 2;
#pragma unroll
    for (int nt = 0; nt < 2; ++nt) {
      const int c = nt * 16 + col;
      v2f t0, t1;
      t0[0] = Wm[kr * NH + c];
      t0[1] = Wm[(kr + 1) * NH + c];
      t1[0] = Wm[NH * NH + kr * NH + c];
      t1[1] = Wm[NH * NH + (kr + 1) * NH + c];
      bM0[ks][nt] = t0;
      bM1[ks][nt] = t1;
    }
  }
  const float vbm0[2] = {bm[col], bm[16 + col]};
  const float vbm1[2] = {bm[NH + col], bm[NH + 16 + col]};

  // ==== Phase C: sm0[h] = sum_n m[n]*tanh(z0@Wm0+bm0)[n,h] ====
  for (int i = 0; i < TPW; ++i) {
    const int ib = (w * TPW + i) * 16;
    v8f c0 = z8(), c1 = z8();
    const float* zA = &lds.z0[(size_t)(ib + col) * NH + khalf * 2];
#pragma unroll
    for (int ks = 0; ks < 8; ++ks) {
      v2f af = *(const v2f*)(zA + ks * 4);
      c0 = wmma4(af, bM0[ks][0], c0);
      c1 = wmma4(af, bM0[ks][1], c1);
    }
    float p0 = 0.f, p1 = 0.f;
#pragma unroll
    for (int r = 0; r < 8; ++r) {
      const float mr = lds.maskf[ib + rowoff + r];
      p0 += mr * tanhf(c0[r] + vbm0[0]);
      p1 += mr * tanhf(c1[r] + vbm0[1]);
    }
    atomicAdd(&lds.sm0[col], p0);
    atomicAdd(&lds.sm0[16 + col], p1);
  }
  __syncthreads();

  // ==== Phase D (wave 0): zbp0 / zbp1 32x32 mat-vecs ====
  if (w == 0) {
    const float invc = 1.f / (lds.cnt + 1e-8f);
    float acc0 = bp[l];
    for (int h = 0; h < NH; ++h) acc0 += (lds.s0[h] * invc) * Wp[h * NH + l];
    lds.zbp0[l] = acc0;
    float acc1 = bp[NH + l];
    for (int h = 0; h < NH; ++h) {
      const float zb1 = (lds.s0[h] + lds.zbp0[h] * lds.sm0[h]) * invc;
      acc1 += zb1 * Wp[NH * NH + h * NH + l];
    }
    lds.zbp1[l] = acc1;
  }
  __syncthreads();

  // ==== Phase E: z2 = z0 + zbp0*modul0 + zbp1*modul1; halo = z2 @ Wh ====
  {
    const float zp00 = lds.zbp0[col], zp01 = lds.zbp0[16 + col];
    const float zp10 = lds.zbp1[col], zp11 = lds.zbp1[16 + col];
    float* ws = lds.ws[w];
    for (int i = 0; i < TPW; ++i) {
      const int ib = (w * TPW + i) * 16;
      v8f c0 = z8(), c1 = z8(), d0 = z8(), d1 = z8();
      const float* zA = &lds.z0[(size_t)(ib + col) * NH + khalf * 2];
#pragma unroll
      for (int ks = 0; ks < 8; ++ks) {  // shared A-fragments, two GEMMs
        v2f af = *(const v2f*)(zA + ks * 4);
        c0 = wmma4(af, bM0[ks][0], c0);
        c1 = wmma4(af, bM0[ks][1], c1);
        d0 = wmma4(af, bM1[ks][0], d0);
        d1 = wmma4(af, bM1[ks][1], d1);
      }
      const float* zrow = &lds.z0[(size_t)ib * NH];
#pragma unroll
      for (int r = 0; r < 8; ++r) {
        const int M = r + rowoff;
        float z2a = zrow[M * NH + col] + zp00 * tanhf(c0[r] + vbm0[0]) +
                    zp10 * tanhf(d0[r] + vbm1[0]);
        float z2b = zrow[M * NH + 16 + col] + zp01 * tanhf(c1[r] + vbm0[1]) +
                    zp11 * tanhf(d1[r] + vbm1[1]);
        ws[M * NH + col] = z2a;
        ws[M * NH + 16 + col] = z2b;
      }
      if (l < 16) {  // head dot-product; Wh reads are uniform -> scalar loads
        float h = 0.f;
        const float4* wr = (const float4*)&ws[l * NH];
#pragma unroll
        for (int q = 0; q < 8; ++q) {
          float4 v4 = wr[q];
          h += v4.x * Wh[4 * q] + v4.y * Wh[4 * q + 1] + v4.z * Wh[4 * q + 2] +
               v4.w * Wh[4 * q + 3];
        }
        lds.halo[ib + l] = h;
      }
    }
  }
  __syncthreads();

  // ==== Phase F: mask / xi gather / coalesced store ====
  {
    const int mid = mids[b];
    const float* xirow = xi + (size_t)mid * NI;
    const float bhead = bh[0];
    for (int n = tid; n < NI; n += THREADS) {
      const float v =
          (lds.maskf[n] != 0.f) ? (lds.halo[n] + bhead + xirow[n]) : -1.0e9f;
      out[(size_t)b * NI + n] = v;
    }
  }
}

extern "C" void kernel_launch(void* const* d_in, const int* in_sizes, int n_in,
                              void* d_out, int out_size, void* d_ws,
                              size_t ws_size, hipStream_t stream) {
  const float* X = (const float*)d_in[0];
  const int* mask = (const int*)d_in[1];
  const int* mids = (const int*)d_in[2];
  const float* We1 = (const float*)d_in[3];
  const float* be1 = (const float*)d_in[4];
  const float* We2 = (const float*)d_in[5];
  const float* be2 = (const float*)d_in[6];
  const float* Wp = (const float*)d_in[7];
  const float* bp = (const float*)d_in[8];
  const float* Wm = (const float*)d_in[9];
  const float* bm = (const float*)d_in[10];
  const float* Wh = (const float*)d_in[11];
  const float* bh = (const float*)d_in[12];
  const float* xi = (const float*)d_in[13];
  float* out = (float*)d_out;

  halo_fused<<<dim3(NB), dim3(THREADS), 0, stream>>>(
      X, mask, mids, We1, be1, We2, be2, Wp, bp, Wm, bm, Wh, bh, xi, out);
}